// RoiPooling_74543452389652
// MI455X (gfx1250) — compile-verified
//
#include <hip/hip_runtime.h>

// ROI crop + bilinear resize of (W:224->56, C:64->56), fp32, per torch semantics.
//   out[b,r,h,p,q] = sum_{i in {4p+1,4p+2}} 0.5 * sum_{j in {j0,j1}} wc_j * x[b, y1+h, x1+i, j]
// Implemented as: async DMA of raw 128-float rows (w=4p+1,4p+2 channel pairs) into LDS
// (GLOBAL_LOAD_ASYNC_TO_LDS_B128, ASYNCcnt), then a 128-K constant-weight GEMM on
// v_wmma_f32_16x16x4_f32 (the 0.5 W-axis average folded into the weight matrix:
// W2[j][q] = W2[64+j][q] = 0.5*Wc[j][q]).

#define POOL   56
#define C_IN   64
#define ROI_H  224
#define W_IN   448
#define R_N    8
#define LDS_STRIDE 132  // floats/row: 528B -> 16B aligned async dest, conflict-free A gathers

typedef __attribute__((ext_vector_type(2))) float v2f;
typedef __attribute__((ext_vector_type(4))) float v4f;
typedef __attribute__((ext_vector_type(8))) float v8f;
typedef __attribute__((ext_vector_type(4))) int   v4i;

#if defined(__gfx1250__) && __has_builtin(__builtin_amdgcn_global_load_async_to_lds_b128)
#define HAVE_ASYNC_LDS 1
#else
#define HAVE_ASYNC_LDS 0
#endif

__device__ __forceinline__ void copy16_g2l(float* lds_dst, const float* gsrc) {
#if HAVE_ASYNC_LDS
    // CDNA5 async copy: global -> LDS without touching the VGPR data path (ASYNCcnt).
    // Builtin signature (from hipcc diagnostic): (v4i AS1*, v4i AS3*, int offset, int cpol).
    __builtin_amdgcn_global_load_async_to_lds_b128(
        (__attribute__((address_space(1))) v4i*)gsrc,
        (__attribute__((address_space(3))) v4i*)lds_dst,
        /*offset=*/0, /*cpol=*/0);
#else
    *(v4f*)lds_dst = *(const v4f*)gsrc;   // fallback: through VGPRs
#endif
}

__device__ __forceinline__ void wait_async_done() {
#if HAVE_ASYNC_LDS
#if __has_builtin(__builtin_amdgcn_s_wait_asynccnt)
    __builtin_amdgcn_s_wait_asynccnt(0);
#else
    asm volatile("s_wait_asynccnt 0" ::: "memory");
#endif
#endif
}

__global__ __launch_bounds__(256) void RoiResizeWmmaKernel(
    const float* __restrict__ x,    // [8,448,448,64]
    const int*   __restrict__ rois, // [8,4] = (x1,y1,w,h)
    float*       __restrict__ out)  // [8,8,224,56,56]
{
    __shared__ float raw[64 * LDS_STRIDE];  // 64 p-rows (padded) x 132 floats = 33792 B

    const int blk = blockIdx.x;            // (b*8 + r)*224 + h
    const int h   = blk % ROI_H;
    const int br  = blk / ROI_H;
    const int r   = br % R_N;
    const int b   = br / R_N;

    const int x1 = rois[r * 4 + 0];
    const int y1 = rois[r * 4 + 1];

    // Base of this (b, y1+h) image row, offset by x1; channels contiguous.
    const float* g = x + (((size_t)b * W_IN + (size_t)(y1 + h)) * W_IN + (size_t)x1) * C_IN;

    const int tid  = threadIdx.x;
    const int lane = tid & 31;             // wave32
    const int wave = tid >> 5;             // 0..7

    // ---------- Phase A: async-stage 56 rows x 128 contiguous floats into LDS ----------
    // Row p holds channels of w=4p+1 (j=0..63) and w=4p+2 (j=64..127): one 512B chunk.
    #pragma unroll
    for (int k = 0; k < 7; ++k) {
        const int p = wave + 8 * k;        // covers 0..55 exactly across 8 waves
        copy16_g2l(&raw[p * LDS_STRIDE + lane * 4],
                   g + (size_t)(4 * p + 1) * C_IN + lane * 4);
    }
    wait_async_done();
    __syncthreads();

    // ---------- Phase B: K=128 constant-weight GEMM on the matrix pipe ----------
    const int halfSel = lane >> 4;         // 0: lanes 0-15, 1: lanes 16-31
    const int ln      = lane & 15;
    const int mtile   = wave & 3;          // A shared between this wave's two N-tiles
    const int nbase   = (wave >> 2) * 2;   // N-tile pair {nbase, nbase+1}

    const float* arow    = &raw[(mtile * 16 + ln) * LDS_STRIDE];
    float*       outBase = out + (size_t)blk * (POOL * POOL);

    // Per-lane channel-resize taps for the two output columns (0.5 W-average folded in).
    int   qcol[2], j0q[2];
    float w0h[2], w1h[2];
    #pragma unroll
    for (int s = 0; s < 2; ++s) {
        const int q = (nbase + s) * 16 + ln;        // 0..63 (>=56 padded)
        qcol[s] = q;
        const float srcc = ((float)q + 0.5f) * (64.0f / 56.0f) - 0.5f;
        const int   j0   = (int)floorf(srcc);
        const float w1   = srcc - (float)j0;
        j0q[s] = j0;
        w1h[s] = (q < POOL) ? 0.5f * w1 : 0.0f;
        w0h[s] = (q < POOL) ? 0.5f * (1.0f - w1) : 0.0f;
    }

    v8f acc0 = {0.f, 0.f, 0.f, 0.f, 0.f, 0.f, 0.f, 0.f};
    v8f acc1 = {0.f, 0.f, 0.f, 0.f, 0.f, 0.f, 0.f, 0.f};

    #pragma unroll
    for (int kk = 0; kk < 32; ++kk) {
        const int k0  = kk * 4 + 2 * halfSel;       // ISA A-layout: lanes<16 K,K+1; >=16 K+2,K+3
        v2f a = *(const v2f*)(arow + k0);           // one ds_load_b64, conflict-free
        const int jm0 = k0 & 63;                    // channel index (both w taps share weights)
        const int jm1 = (k0 + 1) & 63;
        v2f b0, b1;
        b0.x = (jm0 == j0q[0]) ? w0h[0] : ((jm0 == j0q[0] + 1) ? w1h[0] : 0.0f);
        b0.y = (jm1 == j0q[0]) ? w0h[0] : ((jm1 == j0q[0] + 1) ? w1h[0] : 0.0f);
        b1.x = (jm0 == j0q[1]) ? w0h[1] : ((jm0 == j0q[1] + 1) ? w1h[1] : 0.0f);
        b1.y = (jm1 == j0q[1]) ? w0h[1] : ((jm1 == j0q[1] + 1) ? w1h[1] : 0.0f);
#if defined(__gfx1250__) && __has_builtin(__builtin_amdgcn_wmma_f32_16x16x4_f32)
        acc0 = __builtin_amdgcn_wmma_f32_16x16x4_f32(false, a, false, b0, (short)0, acc0, false, false);
        acc1 = __builtin_amdgcn_wmma_f32_16x16x4_f32(false, a, false, b1, (short)0, acc1, false, false);
#else
        acc0[0] += a.x * b0.x + a.y * b0.y;         // compile-safety placeholder only
        acc1[0] += a.x * b1.x + a.y * b1.y;
#endif
    }

    // D layout: VGPR v -> M = v (lanes 0-15) or v+8 (lanes 16-31); N = ln.
    const int pbase = mtile * 16 + 8 * halfSel;
    if (qcol[0] < POOL) {
        #pragma unroll
        for (int v = 0; v < 8; ++v) {
            const int p = pbase + v;
            if (p < POOL) outBase[(size_t)p * POOL + qcol[0]] = acc0[v];
        }
    }
    if (qcol[1] < POOL) {
        #pragma unroll
        for (int v = 0; v < 8; ++v) {
            const int p = pbase + v;
            if (p < POOL) outBase[(size_t)p * POOL + qcol[1]] = acc1[v];
        }
    }
}

extern "C" void kernel_launch(void* const* d_in, const int* in_sizes, int n_in,
                              void* d_out, int out_size, void* d_ws, size_t ws_size,
                              hipStream_t stream) {
    const float* x    = (const float*)d_in[0];
    const int*   rois = (const int*)d_in[1];
    float*       out  = (float*)d_out;

    const int blocks = 8 * R_N * ROI_H;   // 14336 = one block per (b, r, h)
    RoiResizeWmmaKernel<<<blocks, 256, 0, stream>>>(x, rois, out);
}